// HashGridEncoding_51110110822808
// MI455X (gfx1250) — compile-verified
//
#include <hip/hip_runtime.h>
#include <cmath>

// Instant-NGP style hash-grid encoding for MI455X (gfx1250, wave32).
// B=4, N=262144 -> 1,048,576 points; 16 levels; tables 16 x 2^19 x 2 f32 (64MB,
// fits in the 192MB L2). Pure gather workload: no matrix math, so no WMMA --
// the win comes from L2-resident gathers, batched loads, and NT output stores.

#define NUM_LEVELS   16
#define HASHMAP_SIZE (1 << 19)
#define HASH_MASK    0x7FFFF
#define BLOCK        256

typedef float v4f __attribute__((ext_vector_type(4)));  // clang vector: OK for NT builtins
typedef float v2f __attribute__((ext_vector_type(2)));

struct LevelParams {
    int res[NUM_LEVELS];   // int(16 * scale^l), computed on host with libm
    int rr [NUM_LEVELS];   // (res*res) & HASH_MASK
};

__global__ __launch_bounds__(BLOCK)
void HashGridEncoding_51110110822808_kernel(const float* __restrict__ x,
                                            const float* __restrict__ emb,
                                            float* __restrict__ out,
                                            LevelParams P, int npts)
{
    const int p = blockIdx.x * BLOCK + threadIdx.x;
    if (p >= npts) return;

    // ---- load the 3 coordinates (streamed once -> non-temporal) ----
    const float* xp = x + (size_t)3 * (size_t)p;
    const float cx = __builtin_nontemporal_load(xp + 0);
    const float cy = __builtin_nontemporal_load(xp + 1);
    const float cz = __builtin_nontemporal_load(xp + 2);

    // ---- phase 1: compute all 16 hash indices (pure VALU, no divisions) ----
    int idx[NUM_LEVELS];
#pragma unroll
    for (int l = 0; l < NUM_LEVELS; ++l) {
        const int   res = P.res[l];
        const float rf  = (float)res;
        // x*res >= 0, so truncating cvt == floor (saves v_floor_f32 per coord).
        // Result is in [0, res]; fold res -> 0 to match the reference's `% res`
        // (only hit when rounding pushes x*res up to exactly res).
        int c0 = (int)(cx * rf); c0 -= (c0 >= res) ? res : 0;
        int c1 = (int)(cy * rf); c1 -= (c1 >= res) ? res : 0;
        int c2 = (int)(cz * rf); c2 -= (c2 >= res) ? res : 0;
        // H = 2^19: the reference's modular-sum reduces to one AND
        // (c0 <= 2047, c1*res <= 2^22, c2*rr < 2^31 -> no overflow)
        idx[l] = (c0 + c1 * res + ((c2 * P.rr[l]) & HASH_MASK)) & HASH_MASK;
    }

    // ---- phase 2: issue all 16 gathers back-to-back (L2-resident tables),
    //      letting LOADcnt batch them to hide latency ----
    v2f f[NUM_LEVELS];
#pragma unroll
    for (int l = 0; l < NUM_LEVELS; ++l) {
        const v2f* __restrict__ tbl =
            (const v2f*)(emb + (size_t)l * (size_t)HASHMAP_SIZE * 2);
        f[l] = tbl[idx[l]];   // global_load_b64, default RT (keep in L2)
    }

    // ---- phase 3: each thread owns one full 128B line of output; write it
    //      with 8 x b128 non-temporal stores so the 128MB output stream does
    //      not evict the embedding tables from L2 ----
    v4f* __restrict__ op = (v4f*)(out + (size_t)p * (NUM_LEVELS * 2));
#pragma unroll
    for (int q = 0; q < 8; ++q) {
        v4f v;
        v.x = f[2 * q    ].x;  v.y = f[2 * q    ].y;
        v.z = f[2 * q + 1].x;  v.w = f[2 * q + 1].y;
        __builtin_nontemporal_store(v, op + q);
    }
}

extern "C" void kernel_launch(void* const* d_in, const int* in_sizes, int n_in,
                              void* d_out, int out_size, void* d_ws, size_t ws_size,
                              hipStream_t stream) {
    (void)n_in; (void)out_size; (void)d_ws; (void)ws_size;
    const float* x   = (const float*)d_in[0];   // [B,N,3] f32
    const float* emb = (const float*)d_in[1];   // [16, 2^19, 2] f32
    float*       out = (float*)d_out;           // [B,N,32] f32

    const int npts = in_sizes[0] / 3;           // B*N

    // Replicate the reference's resolution schedule with the exact same
    // double-precision expression (same libm as Python's math.exp/log and
    // float.__pow__), so int() truncation boundaries match.
    LevelParams P;
    const double scale = exp((log(2048.0) - log(16.0)) / 15.0);
    for (int l = 0; l < NUM_LEVELS; ++l) {
        const int res = (int)(16.0 * pow(scale, (double)l));
        P.res[l] = res;
        P.rr[l]  = (res * res) & HASH_MASK;
    }

    const int blocks = (npts + BLOCK - 1) / BLOCK;
    HashGridEncoding_51110110822808_kernel<<<blocks, BLOCK, 0, stream>>>(
        x, emb, out, P, npts);
}